// PKGLabelOnehot_67619965108954
// MI455X (gfx1250) — compile-verified
//
#include <hip/hip_runtime.h>
#include <hip/hip_bf16.h>
#include <stdint.h>

// PKG one-hot scatter: out[b,t,c] = probabilities[l] where c == PKG_label[b,l,t],
// later l wins on collision; everything else 0.
//
// Pure store-bandwidth problem: 256 MB of fp32 stores (floor ~11 us @ 23.3 TB/s).
// Strategy: build each 4096-float row in LDS with its FINAL values (3 uniform
// compares per element resolve the l-priority in-register), then stream the row
// to HBM with CDNA5 async LDS->global b128 stores (ASYNCcnt path), so every
// output byte is written exactly once by a wide store.

namespace {
constexpr int kC       = 4096;  // CLASS_DIM_GRAPH
constexpr int kL       = 3;
constexpr int kT       = 128;
constexpr int kThreads = 256;                      // 8 waves (wave32)
constexpr int kVecPT   = kC / 4 / kThreads;        // 4 float4 chunks per thread
}

__global__ __launch_bounds__(kThreads)
void pkg_onehot_async_kernel(const int* __restrict__ labels,   // (B, L, T)
                             const float* __restrict__ probs,  // (L,)
                             float* __restrict__ out)          // (B, T, C)
{
    __shared__ float4 srow[kC / 4];                // 16 KB staged output row

    const int row = blockIdx.x;                    // row = b*T + t
    const int b   = row >> 7;                      // / 128
    const int t   = row & (kT - 1);

    // Block-uniform: the three class indices and probabilities for this (b,t).
    const int   i0 = labels[(b * kL + 0) * kT + t];
    const int   i1 = labels[(b * kL + 1) * kT + t];
    const int   i2 = labels[(b * kL + 2) * kT + t];
    const float p0 = probs[0];
    const float p1 = probs[1];
    const float p2 = probs[2];

    float* __restrict__ rowp = out + (size_t)row * kC;
    const int tid = threadIdx.x;

    // Build the final row in LDS: later l has priority (matches jnp.where chain).
#pragma unroll
    for (int j = 0; j < kVecPT; ++j) {
        const int q  = j * kThreads + tid;         // float4 slot in row
        const int c0 = q * 4;
        float4 w;
        w.x = (c0 + 0 == i2) ? p2 : (c0 + 0 == i1) ? p1 : (c0 + 0 == i0) ? p0 : 0.0f;
        w.y = (c0 + 1 == i2) ? p2 : (c0 + 1 == i1) ? p1 : (c0 + 1 == i0) ? p0 : 0.0f;
        w.z = (c0 + 2 == i2) ? p2 : (c0 + 2 == i1) ? p1 : (c0 + 2 == i0) ? p0 : 0.0f;
        w.w = (c0 + 3 == i2) ? p2 : (c0 + 3 == i1) ? p1 : (c0 + 3 == i0) ? p0 : 0.0f;
        srow[q] = w;                               // ds_store_b128, conflict-free
    }

    // Same-wave DS writes vs async-engine LDS reads are unordered: drain DScnt
    // before handing the row to the async store engine.
    asm volatile("s_wait_dscnt 0" ::: "memory");

    // Stream LDS -> HBM with async b128 stores (tracked by ASYNCcnt).
    // Per instruction each wave writes 512 contiguous bytes.
#pragma unroll
    for (int j = 0; j < kVecPT; ++j) {
        const int q = j * kThreads + tid;
        const uint64_t gaddr = (uint64_t)(uintptr_t)(rowp + (size_t)q * 4);
        const uint32_t laddr = (uint32_t)(uintptr_t)(&srow[q]); // low 32 bits = LDS byte addr
        asm volatile("global_store_async_from_lds_b128 %0, %1, off"
                     :: "v"(gaddr), "v"(laddr)
                     : "memory");
    }

    // S_ENDPGM performs an implicit wait-idle, but make the drain explicit.
    asm volatile("s_wait_asynccnt 0" ::: "memory");
}

extern "C" void kernel_launch(void* const* d_in, const int* in_sizes, int n_in,
                              void* d_out, int out_size, void* d_ws, size_t ws_size,
                              hipStream_t stream)
{
    const int*   labels = (const int*)d_in[0];    // (B, L, T) int
    const float* probs  = (const float*)d_in[1];  // (L,) float32
    float*       out    = (float*)d_out;          // (B, T, C) float32

    const int B    = in_sizes[0] / (kL * kT);     // 128
    const int rows = B * kT;                      // 16384 independent rows

    pkg_onehot_async_kernel<<<rows, kThreads, 0, stream>>>(labels, probs, out);
}